// LightGCN_49469433316104
// MI455X (gfx1250) — compile-verified
//
#include <hip/hip_runtime.h>

// ---------------- problem constants (match reference) ----------------
#define N_USERS  100000
#define N_ITEMS  50000
#define N_TOTAL  (N_USERS + N_ITEMS)
#define FACTORS  64
#define LAYERS   3
#define BATCH    16384

typedef __attribute__((ext_vector_type(2))) float v2f;
typedef __attribute__((ext_vector_type(8))) float v8f;

// ---------------------------------------------------------------------
// init: emb0 = concat(user_w, item_w); acc = emb0   (float4 vectorized)
// ---------------------------------------------------------------------
__global__ void lgcn_init(const float4* __restrict__ uw,
                          const float4* __restrict__ iw,
                          float4* __restrict__ emb,
                          float4* __restrict__ acc) {
  const int total4  = N_TOTAL * FACTORS / 4;
  const int uelems4 = N_USERS * FACTORS / 4;
  for (int i = blockIdx.x * blockDim.x + threadIdx.x; i < total4;
       i += gridDim.x * blockDim.x) {
    float4 v = (i < uelems4) ? uw[i] : iw[i - uelems4];
    emb[i] = v;
    acc[i] = v;
  }
}

__global__ void lgcn_zero(float4* __restrict__ p) {
  const int total4 = N_TOTAL * FACTORS / 4;
  const float4 z = make_float4(0.f, 0.f, 0.f, 0.f);
  for (int i = blockIdx.x * blockDim.x + threadIdx.x; i < total4;
       i += gridDim.x * blockDim.x)
    p[i] = z;
}

__global__ void lgcn_acc_add(float4* __restrict__ acc,
                             const float4* __restrict__ s) {
  const int total4 = N_TOTAL * FACTORS / 4;
  for (int i = blockIdx.x * blockDim.x + threadIdx.x; i < total4;
       i += gridDim.x * blockDim.x) {
    float4 a = acc[i], b = s[i];
    a.x += b.x; a.y += b.y; a.z += b.z; a.w += b.w;
    acc[i] = a;
  }
}

// ---------------------------------------------------------------------
// COO SpMM: one wave32 per nonzero; lane L handles the float2 slice
// [2L, 2L+1] of the 64-wide row. Gather src row (L2-resident, 256B
// coalesced), scale by val, scatter with f32 atomics to dst row.
// ---------------------------------------------------------------------
__global__ void lgcn_spmm(const float* __restrict__ vals,
                          const int*   __restrict__ rows,
                          const int*   __restrict__ cols,
                          const float* __restrict__ src,
                          float*       __restrict__ dst,
                          int nnz) {
  const int lane   = threadIdx.x & 31;
  const int wave   = (blockIdx.x * blockDim.x + threadIdx.x) >> 5;
  const int nwaves = (gridDim.x * blockDim.x) >> 5;
  for (int e = wave; e < nnz; e += nwaves) {
    const float v = vals[e];
    const int   r = rows[e];
    const int   c = cols[e];
    const float2 x = ((const float2*)(src + (size_t)c * FACTORS))[lane];
    float* d = dst + (size_t)r * FACTORS + 2 * lane;
    atomicAdd(d,     x.x * v);
    atomicAdd(d + 1, x.y * v);
  }
}

// ---------------------------------------------------------------------
// Batched dot via f32 WMMA: each wave handles 16 (user,item) pairs.
// D = U (16x64) * I^T (64x16) accumulated by 16 steps of
// V_WMMA_F32_16X16X4_F32; diag(D) = the 16 dot products.
//
// A-fragment layout (ISA 7.12.2, 32-bit A 16x4): lane m (0-15) holds
// A[m][k0+0], A[m][k0+1]; lane 16+m holds A[m][k0+2], A[m][k0+3].
// B-fragment mirrored with N striped across lanes -> identical per-lane
// float2 load pattern from the item row.
// C/D layout: VGPR j -> M=j (lanes 0-15) / M=8+j (lanes 16-31), N=lane%16.
// ---------------------------------------------------------------------
__global__ void lgcn_dot_wmma(const float* __restrict__ emb,   // acc [N,64]
                              const int*   __restrict__ users,
                              const int*   __restrict__ items,
                              float*       __restrict__ out) {
  const int wave = (blockIdx.x * blockDim.x + threadIdx.x) >> 5;
  const int lane = threadIdx.x & 31;
  const int base = wave * 16;           // grid sized so base < BATCH always

  const int m  = lane & 15;
  const int hi = lane >> 4;             // 0: K=0,1   1: K=2,3

  const float* pu = emb + (size_t)users[base + m] * FACTORS;
  const float* pi = emb + ((size_t)N_USERS + (size_t)items[base + m]) * FACTORS;

  v8f c = {};
#pragma unroll
  for (int k = 0; k < FACTORS; k += 4) {
    const float2 a2 = *(const float2*)(pu + k + 2 * hi);
    const float2 b2 = *(const float2*)(pi + k + 2 * hi);
    v2f a; a[0] = a2.x; a[1] = a2.y;
    v2f b; b[0] = b2.x; b[1] = b2.y;
    c = __builtin_amdgcn_wmma_f32_16x16x4_f32(
        /*neg_a=*/false, a, /*neg_b=*/false, b,
        /*c_mod=*/(short)0, c, /*reuse_a=*/false, /*reuse_b=*/false);
  }

  // diagonal extraction: D[j][j] lives in vgpr j on lane j (j<8),
  // D[8+j][8+j] in vgpr j on lane 24+j; the loop reads lane 16+j (D[8+j][j]).
  float d[16];
#pragma unroll
  for (int j = 0; j < 8; ++j) {
    d[j]     = __shfl(c[j], j, 32);
    d[j + 8] = __shfl(c[j], j + 16, 32);
  }

  if (lane == 0) {
    const float s = 1.0f / ((LAYERS + 1) * (LAYERS + 1));  // (1/4)*(1/4)
    float4* o = (float4*)(out + base);
    o[0] = make_float4(d[0] * s,  d[1] * s,  d[2] * s,  d[3] * s);
    o[1] = make_float4(d[4] * s,  d[5] * s,  d[6] * s,  d[7] * s);
    o[2] = make_float4(d[8] * s,  d[9] * s,  d[10] * s, d[11] * s);
    o[3] = make_float4(d[12] * s, d[13] * s, d[14] * s, d[15] * s);
  }
}

// ---------------------------------------------------------------------
extern "C" void kernel_launch(void* const* d_in, const int* in_sizes, int n_in,
                              void* d_out, int out_size, void* d_ws, size_t ws_size,
                              hipStream_t stream) {
  const float* user_w   = (const float*)d_in[0];
  const float* item_w   = (const float*)d_in[1];
  const float* adj_vals = (const float*)d_in[2];
  const int*   adj_row  = (const int*)d_in[3];
  const int*   adj_col  = (const int*)d_in[4];
  const int*   users    = (const int*)d_in[5];
  const int*   items    = (const int*)d_in[6];
  const int    nnz      = in_sizes[2];
  float*       out      = (float*)d_out;

  const size_t NT = (size_t)N_TOTAL * FACTORS;   // 9.6M floats = 38.4 MB
  float* bufA = (float*)d_ws;                    // layer input
  float* bufB = bufA + NT;                       // layer output
  float* acc  = bufB + NT;                       // running layer sum

  const int TB = 256;
  const int vec_blocks = (int)((NT / 4 + TB - 1) / TB);

  lgcn_init<<<vec_blocks, TB, 0, stream>>>(
      (const float4*)user_w, (const float4*)item_w,
      (float4*)bufA, (float4*)acc);

  float* prev = bufA;
  float* next = bufB;
  const int spmm_blocks = 4096;                  // 32768 waves, grid-stride
  for (int l = 0; l < LAYERS; ++l) {
    lgcn_zero<<<vec_blocks, TB, 0, stream>>>((float4*)next);
    lgcn_spmm<<<spmm_blocks, TB, 0, stream>>>(adj_vals, adj_row, adj_col,
                                              prev, next, nnz);
    lgcn_acc_add<<<vec_blocks, TB, 0, stream>>>((float4*)acc,
                                                (const float4*)next);
    float* t = prev; prev = next; next = t;
  }

  // 16 dots per wave -> BATCH/16 = 1024 waves -> 32768 threads
  const int dot_blocks = (BATCH / 16) * 32 / TB; // 128
  lgcn_dot_wmma<<<dot_blocks, TB, 0, stream>>>(acc, users, items, out);
}